// InductiveMDGCNLayer_14078902796410
// MI455X (gfx1250) — compile-verified
//
#include <hip/hip_runtime.h>

// MI455X (gfx1250) implementation of the multi-hop GCN layer.
//
// Factorization: out = relu( sum_k d_k ⊙ [ A_k (d_k⊙XW_k) + α E (Eᵀ (d_k⊙XW_k)) ] ),
//                deg_k = rowsum(A_k) + α E (colsum E).
// Dominant cost: 3x (4096x4096 @ 4096x64) fp32 GEMM streaming A (201 MB) from HBM,
// done with V_WMMA_F32_16X16X4_F32. B-operand (yT) is staged into LDS with
// double-buffered GLOBAL_LOAD_ASYNC_TO_LDS_B128 (ASYNCcnt) and read bank-conflict-free.

typedef __attribute__((ext_vector_type(2))) float v2f;
typedef __attribute__((ext_vector_type(8))) float v8f;

#define N_NODES 4096
#define F_DIM   128
#define D_DIM   64
#define NHOPS   3
#define KSPLIT  4
#define JCHUNK  256                 // K-chunk staged in LDS per double-buffer slot
#define LDS_ROW (JCHUNK + 4)        // +4 floats pad -> ds_load_b64 hits all 64 banks

__device__ __forceinline__ void async_ld_b128(unsigned lds_off, unsigned long long gaddr) {
  asm volatile("global_load_async_to_lds_b128 %0, %1, off"
               :
               : "v"(lds_off), "v"(gaddr)
               : "memory");
}
__device__ __forceinline__ void wait_async0() {
  asm volatile("s_wait_asynccnt 0x0" ::: "memory");
}

// ---------------- K1: E = X @ W_emb ; XW[k] = X @ W_hops[k] ----------------
__global__ void k_proj(const float* __restrict__ X, const float* __restrict__ W_emb,
                       const float* __restrict__ W_hops,
                       float* __restrict__ E, float* __restrict__ XW) {
  int id  = blockIdx.x * blockDim.x + threadIdx.x;   // 4 * 4096 * 64 threads
  int w   = id >> 18;                                // which weight matrix (0 = emb)
  int rem = id & ((1 << 18) - 1);
  int i   = rem >> 6;
  int d   = rem & 63;
  const float* W = (w == 0) ? W_emb : (W_hops + (size_t)(w - 1) * F_DIM * D_DIM);
  const float* x = X + (size_t)i * F_DIM;
  float acc = 0.f;
#pragma unroll 8
  for (int f = 0; f < F_DIM; ++f) acc = fmaf(x[f], W[f * D_DIM + d], acc);
  if (w == 0) E[(size_t)i * D_DIM + d] = acc;
  else        XW[(size_t)(w - 1) * N_NODES * D_DIM + (size_t)i * D_DIM + d] = acc;
}

// ---------------- K2: R[k][i] = rowsum(A[k]) (one wave per row) ----------------
// Regular (RT) loads: fills L2 so the GEMM pass can re-hit recently streamed hops.
__global__ void k_rowsum(const float* __restrict__ A, float* __restrict__ R) {
  int gtid = blockIdx.x * blockDim.x + threadIdx.x;
  int wave = gtid >> 5;            // 0 .. 3*4096-1
  int lane = threadIdx.x & 31;
  const float* row = A + (size_t)wave * N_NODES;
  float acc = 0.f;
  for (int j = lane * 4; j < N_NODES; j += 32 * 4) {
    float4 v = *(const float4*)(row + j);
    acc += v.x + v.y + v.z + v.w;
  }
#pragma unroll
  for (int off = 16; off > 0; off >>= 1) acc += __shfl_xor(acc, off, 32);
  if (lane == 0) R[wave] = acc;
}

// ---------------- K3: s[e] = colsum(E) ----------------
__global__ void k_colsum(const float* __restrict__ E, float* __restrict__ s) {
  int e = threadIdx.x;             // 64 threads
  float acc = 0.f;
  for (int i = 0; i < N_NODES; ++i) acc += E[(size_t)i * D_DIM + e];
  s[e] = acc;
}

// ------- K4: t = E@s ; d_k = rsqrt(R_k + a t) ; yT_k[d][i] = d_k[i]*XW_k[i][d] -------
__global__ void k_norm(const float* __restrict__ E, const float* __restrict__ XW,
                       const float* __restrict__ R, const float* __restrict__ s,
                       const float* __restrict__ alpha_p,
                       float* __restrict__ dk, float* __restrict__ yT) {
  int i = blockIdx.x * blockDim.x + threadIdx.x;     // 4096 threads
  float alpha = *alpha_p;
  const float* e = E + (size_t)i * D_DIM;
  float t = 0.f;
#pragma unroll
  for (int c = 0; c < D_DIM; ++c) t = fmaf(e[c], s[c], t);
  for (int k = 0; k < NHOPS; ++k) {
    float deg = R[k * N_NODES + i] + alpha * t;
    float d   = rsqrtf(deg);
    dk[k * N_NODES + i] = d;
    const float* xw = XW + (size_t)k * N_NODES * D_DIM + (size_t)i * D_DIM;
    float* yt = yT + (size_t)k * N_NODES * D_DIM;
    for (int c = 0; c < D_DIM; ++c) yt[(size_t)c * N_NODES + i] = d * xw[c];
  }
}

// ------- K5: ct_k[d][e] = alpha * sum_i E[i][e] * yT_k[d][i]  (= alpha*(Eᵀ y_k)ᵀ) -------
__global__ void k_lowrank(const float* __restrict__ E, const float* __restrict__ yT,
                          const float* __restrict__ alpha_p, float* __restrict__ ct) {
  int id = blockIdx.x * blockDim.x + threadIdx.x;    // 3*64*64 threads
  int k  = id >> 12;
  int de = id & 4095;
  int d  = de >> 6;
  int e  = de & 63;
  const float* yt = yT + (size_t)k * N_NODES * D_DIM + (size_t)d * N_NODES;
  float acc = 0.f;
  for (int i = 0; i < N_NODES; ++i) acc = fmaf(E[(size_t)i * D_DIM + e], yt[i], acc);
  ct[k * D_DIM * D_DIM + d * D_DIM + e] = (*alpha_p) * acc;
}

// ---------------- K6: main WMMA GEMM: acc += d_k ⊙ ([A_k | E] @ [y_k ; α c_k]) ----------------
// 8 waves/block share an LDS-staged yT K-chunk (async double buffer).
__global__ void __launch_bounds__(256)
k_gemm(const float* __restrict__ A, const float* __restrict__ E,
       const float* __restrict__ yT, const float* __restrict__ ct,
       const float* __restrict__ dk, float* __restrict__ accb) {
  __shared__ float sy[2][D_DIM * LDS_ROW];           // 2 x 65 KB double buffer

  const int k    = (NHOPS - 1) - blockIdx.y;         // hop 2 first: best L2 reuse after rowsum
  const int z    = blockIdx.z;                       // K-split chunk
  const int tid  = threadIdx.x;
  const int wave = tid >> 5;
  const int lane = tid & 31;
  const int m    = lane & 15;                        // row-in-tile (A) / col-in-tile (B,C,D)
  const int h    = lane >> 4;                        // half-wave selector
  const int i0   = (blockIdx.x * 8 + wave) * 16;     // output row block

  const int jlen = N_NODES / KSPLIT;                 // 1024
  const int j0   = z * jlen;

  // A operand: lane holds A[i0+m][j + 2h + {0,1}] (streamed, last-use -> non-temporal)
  const float* ap = A + (size_t)k * N_NODES * N_NODES + (size_t)(i0 + m) * N_NODES + 2 * h + j0;
  const float* yb = yT + (size_t)k * N_NODES * D_DIM + j0;   // 64 x jlen window

  // Staging assignment: 64 rows x JCHUNK floats = 4096 b128 transfers / 256 threads = 16 each
  const int srow0 = tid >> 6;                        // 0..3
  const int scol  = (tid & 63) << 2;                 // float col (steps of 4)
  const unsigned lds_base = (unsigned)(uintptr_t)&sy[0][0];

  // B-read LDS float indices per column tile; pad makes banks = 4m+2h (conflict-free)
  const int b0i = (0 * 16 + m) * LDS_ROW + 2 * h;
  const int b1i = (1 * 16 + m) * LDS_ROW + 2 * h;
  const int b2i = (2 * 16 + m) * LDS_ROW + 2 * h;
  const int b3i = (3 * 16 + m) * LDS_ROW + 2 * h;

  v8f acc0 = {}, acc1 = {}, acc2 = {}, acc3 = {};

  // prologue: stage chunk 0 into buffer 0
#pragma unroll
  for (int it = 0; it < 16; ++it) {
    int row = it * 4 + srow0;
    unsigned loff = lds_base + (unsigned)((row * LDS_ROW + scol) * 4);
    async_ld_b128(loff, (unsigned long long)(uintptr_t)(yb + (size_t)row * N_NODES + scol));
  }
  wait_async0();
  __syncthreads();

  for (int jc = 0; jc < jlen; jc += JCHUNK) {
    const int buf = (jc / JCHUNK) & 1;
    if (jc + JCHUNK < jlen) {                        // prefetch next chunk into other buffer
      unsigned bufoff = (buf ^ 1) ? (unsigned)(D_DIM * LDS_ROW * 4) : 0u;
#pragma unroll
      for (int it = 0; it < 16; ++it) {
        int row = it * 4 + srow0;
        unsigned loff = lds_base + bufoff + (unsigned)((row * LDS_ROW + scol) * 4);
        async_ld_b128(loff,
          (unsigned long long)(uintptr_t)(yb + (size_t)row * N_NODES + jc + JCHUNK + scol));
      }
    }
    const float* apc = ap + jc;
    const float* sb  = &sy[buf][0];
#pragma unroll 4
    for (int j = 0; j < JCHUNK; j += 4) {
      v2f a  = __builtin_nontemporal_load((const v2f*)(apc + j));
      v2f q0 = *(const v2f*)(sb + b0i + j);
      v2f q1 = *(const v2f*)(sb + b1i + j);
      v2f q2 = *(const v2f*)(sb + b2i + j);
      v2f q3 = *(const v2f*)(sb + b3i + j);
      acc0 = __builtin_amdgcn_wmma_f32_16x16x4_f32(false, a, false, q0, (short)0, acc0, false, false);
      acc1 = __builtin_amdgcn_wmma_f32_16x16x4_f32(false, a, false, q1, (short)0, acc1, false, false);
      acc2 = __builtin_amdgcn_wmma_f32_16x16x4_f32(false, a, false, q2, (short)0, acc2, false, false);
      acc3 = __builtin_amdgcn_wmma_f32_16x16x4_f32(false, a, false, q3, (short)0, acc3, false, false);
    }
    wait_async0();       // my async writes for next chunk landed
    __syncthreads();     // everyone done reading buf & done staging buf^1
  }

  if (z == 0) {
    // Low-rank augmentation: 64 extra K-steps, A-operand = E rows, B = α(Eᵀy)ᵀ
    const float* ep = E + (size_t)(i0 + m) * D_DIM + 2 * h;
    const float* cb = ct + (size_t)k * D_DIM * D_DIM;
    const float* c0 = cb + (0 * 16 + m) * D_DIM + 2 * h;
    const float* c1 = cb + (1 * 16 + m) * D_DIM + 2 * h;
    const float* c2 = cb + (2 * 16 + m) * D_DIM + 2 * h;
    const float* c3 = cb + (3 * 16 + m) * D_DIM + 2 * h;
#pragma unroll
    for (int j = 0; j < D_DIM; j += 4) {
      v2f a  = *(const v2f*)(ep + j);
      v2f q0 = *(const v2f*)(c0 + j);
      v2f q1 = *(const v2f*)(c1 + j);
      v2f q2 = *(const v2f*)(c2 + j);
      v2f q3 = *(const v2f*)(c3 + j);
      acc0 = __builtin_amdgcn_wmma_f32_16x16x4_f32(false, a, false, q0, (short)0, acc0, false, false);
      acc1 = __builtin_amdgcn_wmma_f32_16x16x4_f32(false, a, false, q1, (short)0, acc1, false, false);
      acc2 = __builtin_amdgcn_wmma_f32_16x16x4_f32(false, a, false, q2, (short)0, acc2, false, false);
      acc3 = __builtin_amdgcn_wmma_f32_16x16x4_f32(false, a, false, q3, (short)0, acc3, false, false);
    }
  }

  // Epilogue: left-scale rows by d_k and accumulate (atomic across k and K-chunks).
  const float* dkp = dk + k * N_NODES + i0;
  float* outp = accb + (size_t)i0 * D_DIM;
#pragma unroll
  for (int v = 0; v < 8; ++v) {
    int row = v + 8 * h;                       // C/D layout: M = vgpr + 8*(lane>>4)
    float ds = dkp[row];
    atomicAdd(outp + (size_t)row * D_DIM +  0 + m, ds * acc0[v]);
    atomicAdd(outp + (size_t)row * D_DIM + 16 + m, ds * acc1[v]);
    atomicAdd(outp + (size_t)row * D_DIM + 32 + m, ds * acc2[v]);
    atomicAdd(outp + (size_t)row * D_DIM + 48 + m, ds * acc3[v]);
  }
}

// ---------------- K7: out = relu(acc) ----------------
__global__ void k_relu(const float* __restrict__ accb, float* __restrict__ out) {
  int id = blockIdx.x * blockDim.x + threadIdx.x;
  out[id] = fmaxf(accb[id], 0.f);
}

extern "C" void kernel_launch(void* const* d_in, const int* in_sizes, int n_in,
                              void* d_out, int out_size, void* d_ws, size_t ws_size,
                              hipStream_t stream) {
  const float* X       = (const float*)d_in[0];
  const float* A       = (const float*)d_in[1];
  const float* W_emb   = (const float*)d_in[2];
  const float* W_hops  = (const float*)d_in[3];
  const float* alpha_p = (const float*)d_in[4];
  float* out = (float*)d_out;

  // Workspace layout (floats): ~8.2 MB total
  float* ws   = (float*)d_ws;
  float* E    = ws;                          // 4096*64          = 262144
  float* XW   = E    + 262144;               // 3*4096*64        = 786432
  float* yT   = XW   + 786432;               // 3*64*4096        = 786432
  float* ct   = yT   + 786432;               // 3*64*64          = 12288
  float* R    = ct   + 12288;                // 3*4096           = 12288
  float* dk   = R    + 12288;                // 3*4096           = 12288
  float* s    = dk   + 12288;                // 64
  float* accb = s    + 64;                   // 4096*64          = 262144

  hipMemsetAsync(accb, 0, (size_t)262144 * sizeof(float), stream);

  k_proj   <<<dim3((4 * N_NODES * D_DIM) / 256), 256, 0, stream>>>(X, W_emb, W_hops, E, XW);
  k_rowsum <<<dim3((NHOPS * N_NODES) / 8), 256, 0, stream>>>(A, R);
  k_colsum <<<dim3(1), 64, 0, stream>>>(E, s);
  k_norm   <<<dim3(N_NODES / 256), 256, 0, stream>>>(E, XW, R, s, alpha_p, dk, yT);
  k_lowrank<<<dim3((NHOPS * D_DIM * D_DIM) / 256), 256, 0, stream>>>(E, yT, alpha_p, ct);
  k_gemm   <<<dim3(N_NODES / 128, NHOPS, KSPLIT), 256, 0, stream>>>(A, E, yT, ct, dk, accb);
  k_relu   <<<dim3((N_NODES * D_DIM) / 256), 256, 0, stream>>>(accb, out);
}